// CausalGCN_8340826488977
// MI455X (gfx1250) — compile-verified
//
#include <hip/hip_runtime.h>
#include <math.h>

#define NN 20000
#define NE 160000
#define NG 500
#define NG_PAD 512
#define FIN 128
#define HID 512
#define NCLS 10
#define NLAYERS 3

typedef __attribute__((ext_vector_type(2))) float v2f;
typedef __attribute__((ext_vector_type(8))) float v8f;

static __device__ __forceinline__ float warp_sum(float v) {
#pragma unroll
  for (int off = 16; off; off >>= 1) v += __shfl_xor(v, off, 32);
  return v;
}

// ---------------- WMMA fp32 GEMM: C[M,512] = A[M,K] @ B[K,512] (+bias)(+relu)
// One wave computes a 16x32 tile of C (two 16x16 WMMA accumulator chains that
// share the A fragment). K and N are compile-time so every A/B load lowers to
// an immediate-offset global_load; A loads are unconditional (caller pads A's
// allocation to a multiple of 16 rows; garbage A rows only yield garbage D
// rows which the predicated store discards). fuse: 0=none, 1=+bias, 2=+bias,relu
template <int K>
__global__ void wmma_gemm_f32(const float* __restrict__ A, const float* __restrict__ B,
                              const float* __restrict__ bias, float* __restrict__ C,
                              int M, int fuse) {
  constexpr int N = 512;
  int wid  = (int)((blockIdx.x * (unsigned)blockDim.x + threadIdx.x) >> 5);
  int lane = threadIdx.x & 31;
  constexpr int nt2 = N / 32;              // double-tiles along N
  int mt = (M + 15) >> 4;
  if (wid >= mt * nt2) return;             // wave-uniform: EXEC stays all-ones
  int tm = wid / nt2, tn = wid - tm * nt2;
  int row0 = tm << 4, col0 = tn << 5;
  int m  = lane & 15;
  int kh = (lane >> 4) << 1;               // 0 for lanes 0-15, 2 for lanes 16-31

  v8f acc0 = {}, acc1 = {};
  const float* Ap = A + (size_t)(row0 + m) * K + kh;
  const float* Bp = B + (size_t)kh * N + col0 + m;

#pragma unroll 8
  for (int k0 = 0; k0 < K; k0 += 4) {
    v2f a, b0, b1;
    a.x  = Ap[k0];
    a.y  = Ap[k0 + 1];
    b0.x = Bp[(size_t)k0 * N];
    b0.y = Bp[(size_t)(k0 + 1) * N];
    b1.x = Bp[(size_t)k0 * N + 16];
    b1.y = Bp[(size_t)(k0 + 1) * N + 16];
    acc0 = __builtin_amdgcn_wmma_f32_16x16x4_f32(false, a, false, b0, (short)0, acc0,
                                                 false, false);
    acc1 = __builtin_amdgcn_wmma_f32_16x16x4_f32(false, a, false, b1, (short)0, acc1,
                                                 false, false);
  }

  int cn0 = col0 + m;
  int cn1 = cn0 + 16;
  int rb  = row0 + ((lane >> 4) << 3);     // lanes 16-31 hold rows +8
  float bv0 = (fuse && bias) ? bias[cn0] : 0.f;
  float bv1 = (fuse && bias) ? bias[cn1] : 0.f;
#pragma unroll
  for (int j = 0; j < 8; ++j) {
    int r = rb + j;
    if (r < M) {
      float o0 = acc0[j] + bv0;
      float o1 = acc1[j] + bv1;
      if (fuse == 2) { o0 = fmaxf(o0, 0.f); o1 = fmaxf(o1, 0.f); }
      C[(size_t)r * N + cn0] = o0;
      C[(size_t)r * N + cn1] = o1;
    }
  }
}

// ---------------- BatchNorm (training mode, gamma=1, beta=1e-4) -------------
__global__ void colsum_kernel(const float* __restrict__ in, int n, int f,
                              float* __restrict__ stats) {
  int c = threadIdx.x;                       // blockDim.x == f
  float s = 0.f, ss = 0.f;
  for (int r = blockIdx.x; r < n; r += gridDim.x) {
    float v = in[(size_t)r * f + c];
    s += v; ss += v * v;
  }
  atomicAdd(&stats[c], s);
  atomicAdd(&stats[f + c], ss);
}

__global__ void bn_apply_kernel(const float* __restrict__ in, float* __restrict__ out,
                                int n, int f, const float* __restrict__ stats) {
  size_t i = (size_t)blockIdx.x * blockDim.x + threadIdx.x;
  size_t tot = (size_t)n * f;
  if (i >= tot) return;
  int c = (int)(i % f);
  float inv_n = 1.f / (float)n;
  float m = stats[c] * inv_n;
  float var = stats[f + c] * inv_n - m * m;
  out[i] = (in[i] - m) * rsqrtf(var + 1e-5f) + 1e-4f;
}

// ---------------- GCN symmetric normalization -------------------------------
__global__ void fill_kernel(float* __restrict__ p, float v, int n) {
  int i = blockIdx.x * blockDim.x + threadIdx.x;
  if (i < n) p[i] = v;
}

__global__ void deg_accum_kernel(const int* __restrict__ row, const float* __restrict__ w,
                                 float* __restrict__ deg, int E) {
  int e = blockIdx.x * blockDim.x + threadIdx.x;
  if (e >= E) return;
  atomicAdd(&deg[row[e]], w ? w[e] : 1.f);
}

__global__ void dis_selfn_kernel(float* __restrict__ deg, float* __restrict__ selfn, int n) {
  int i = blockIdx.x * blockDim.x + threadIdx.x;
  if (i >= n) return;
  float d = deg[i];
  float dis = d > 0.f ? rsqrtf(d) : 0.f;
  deg[i] = dis;                // deg buffer becomes dis
  selfn[i] = dis * dis;        // self-loop norm (weight 1)
}

__global__ void edge_norm_kernel(const int* __restrict__ row, const int* __restrict__ col,
                                 const float* __restrict__ w, const float* __restrict__ dis,
                                 float* __restrict__ nrm, int E) {
  int e = blockIdx.x * blockDim.x + threadIdx.x;
  if (e >= E) return;
  nrm[e] = dis[row[e]] * (w ? w[e] : 1.f) * dis[col[e]];
}

// ---------------- Sparse propagation: out[r] = selfn[r]*h[r] + sum nrm*h[c] --
__global__ void prop_init_kernel(const float* __restrict__ h, const float* __restrict__ selfn,
                                 float* __restrict__ out) {
  const int CH = HID / 4;
  size_t i = (size_t)blockIdx.x * blockDim.x + threadIdx.x;
  if (i >= (size_t)NN * CH) return;
  int r = (int)(i / CH);
  float s = selfn[r];
  float4 v = ((const float4*)h)[i];
  float4 o; o.x = s * v.x; o.y = s * v.y; o.z = s * v.z; o.w = s * v.w;
  ((float4*)out)[i] = o;
}

__global__ void prop_edge_kernel(const int* __restrict__ row, const int* __restrict__ col,
                                 const float* __restrict__ nrm, const float* __restrict__ h,
                                 float* __restrict__ out) {
  const int CH = HID / 4;
  size_t i = (size_t)blockIdx.x * blockDim.x + threadIdx.x;
  if (i >= (size_t)NE * CH) return;
  int e = (int)(i / CH);
  int q = (int)(i % CH);
  float w = nrm[e];
  float4 v = ((const float4*)(h + (size_t)col[e] * HID))[q];
  float* o = out + (size_t)row[e] * HID + q * 4;
  atomicAdd(o + 0, w * v.x);
  atomicAdd(o + 1, w * v.y);
  atomicAdd(o + 2, w * v.z);
  atomicAdd(o + 3, w * v.w);
}

__global__ void bias_relu_kernel(float* __restrict__ x, const float* __restrict__ b, int n) {
  size_t i = (size_t)blockIdx.x * blockDim.x + threadIdx.x;
  if (i >= (size_t)n * HID) return;
  int c = (int)(i % HID);
  x[i] = fmaxf(x[i] + b[c], 0.f);
}

// ---------------- Attention: per-node dots (wave per node) -------------------
__global__ void node_att_kernel(const float* __restrict__ x, const float* __restrict__ naW,
                                const float* __restrict__ naB, const float* __restrict__ eaW,
                                float* __restrict__ na0, float* __restrict__ na1,
                                float* __restrict__ u0, float* __restrict__ u1,
                                float* __restrict__ v0, float* __restrict__ v1) {
  int wid  = (int)((blockIdx.x * (unsigned)blockDim.x + threadIdx.x) >> 5);
  int lane = threadIdx.x & 31;
  if (wid >= NN) return;
  const float* xr = x + (size_t)wid * HID;
  float s0 = 0.f, s1 = 0.f, s2 = 0.f, s3 = 0.f, s4 = 0.f, s5 = 0.f;
  for (int f = lane; f < HID; f += 32) {
    float xv = xr[f];
    s0 += xv * naW[f * 2 + 0];
    s1 += xv * naW[f * 2 + 1];
    s2 += xv * eaW[f * 2 + 0];
    s3 += xv * eaW[f * 2 + 1];
    s4 += xv * eaW[(HID + f) * 2 + 0];
    s5 += xv * eaW[(HID + f) * 2 + 1];
  }
  s0 = warp_sum(s0); s1 = warp_sum(s1); s2 = warp_sum(s2);
  s3 = warp_sum(s3); s4 = warp_sum(s4); s5 = warp_sum(s5);
  if (lane == 0) {
    float l0 = s0 + naB[0], l1 = s1 + naB[1];
    float mx = fmaxf(l0, l1);
    float e0 = expf(l0 - mx), e1 = expf(l1 - mx);
    float inv = 1.f / (e0 + e1);
    na0[wid] = e0 * inv; na1[wid] = e1 * inv;
    u0[wid] = s2; u1[wid] = s3; v0[wid] = s4; v1[wid] = s5;
  }
}

__global__ void edge_att_kernel(const int* __restrict__ row, const int* __restrict__ col,
                                const float* __restrict__ u0, const float* __restrict__ u1,
                                const float* __restrict__ v0, const float* __restrict__ v1,
                                const float* __restrict__ eaB,
                                float* __restrict__ ea0, float* __restrict__ ea1) {
  int e = blockIdx.x * blockDim.x + threadIdx.x;
  if (e >= NE) return;
  int r = row[e], c = col[e];
  float l0 = u0[r] + v0[c] + eaB[0];
  float l1 = u1[r] + v1[c] + eaB[1];
  float mx = fmaxf(l0, l1);
  float x0 = expf(l0 - mx), x1 = expf(l1 - mx);
  float inv = 1.f / (x0 + x1);
  ea0[e] = x0 * inv; ea1[e] = x1 * inv;
}

__global__ void scale_split_kernel(const float* __restrict__ x, const float* __restrict__ na0,
                                   const float* __restrict__ na1, float* __restrict__ xc,
                                   float* __restrict__ xo) {
  size_t i = (size_t)blockIdx.x * blockDim.x + threadIdx.x;
  if (i >= (size_t)NN * HID) return;
  int r = (int)(i / HID);
  float xv = x[i];
  xc[i] = na0[r] * xv;
  xo[i] = na1[r] * xv;
}

// ---------------- Pooling, perm-add, final classifier ------------------------
__global__ void pool_kernel(const float* __restrict__ x, const int* __restrict__ batch,
                            float* __restrict__ out) {
  const int CH = HID / 4;
  size_t i = (size_t)blockIdx.x * blockDim.x + threadIdx.x;
  if (i >= (size_t)NN * CH) return;
  int r = (int)(i / CH);
  int q = (int)(i % CH);
  int g = batch[r];
  float4 v = ((const float4*)(x + (size_t)r * HID))[q];
  float* o = out + (size_t)g * HID + q * 4;
  atomicAdd(o + 0, v.x); atomicAdd(o + 1, v.y);
  atomicAdd(o + 2, v.z); atomicAdd(o + 3, v.w);
}

__global__ void perm_add_kernel(const float* __restrict__ pc, const float* __restrict__ po,
                                const int* __restrict__ perm, float* __restrict__ out) {
  int i = blockIdx.x * blockDim.x + threadIdx.x;
  if (i >= NG * HID) return;
  int g = i / HID, f = i - g * HID;
  out[i] = pc[(size_t)perm[g] * HID + f] + po[i];
}

__global__ void fc2_logsoftmax_kernel(const float* __restrict__ x, const float* __restrict__ W,
                                      const float* __restrict__ b, float* __restrict__ out) {
  int wid  = (int)((blockIdx.x * (unsigned)blockDim.x + threadIdx.x) >> 5);
  int lane = threadIdx.x & 31;
  if (wid >= NG) return;
  const float* xr = x + (size_t)wid * HID;
  float s[NCLS];
#pragma unroll
  for (int c = 0; c < NCLS; ++c) s[c] = 0.f;
  for (int f = lane; f < HID; f += 32) {
    float xv = xr[f];
#pragma unroll
    for (int c = 0; c < NCLS; ++c) s[c] += xv * W[f * NCLS + c];
  }
#pragma unroll
  for (int c = 0; c < NCLS; ++c) s[c] = warp_sum(s[c]);
  if (lane == 0) {
    float mx = -1e30f;
#pragma unroll
    for (int c = 0; c < NCLS; ++c) { s[c] += b[c]; mx = fmaxf(mx, s[c]); }
    float se = 0.f;
#pragma unroll
    for (int c = 0; c < NCLS; ++c) se += expf(s[c] - mx);
    float lse = mx + logf(se);
#pragma unroll
    for (int c = 0; c < NCLS; ++c) out[(size_t)wid * NCLS + c] = s[c] - lse;
  }
}

// ============================================================================
extern "C" void kernel_launch(void* const* d_in, const int* in_sizes, int n_in,
                              void* d_out, int out_size, void* d_ws, size_t ws_size,
                              hipStream_t stream) {
  (void)in_sizes; (void)n_in; (void)out_size; (void)ws_size;

  const float* x_in  = (const float*)d_in[0];
  const int*   ei    = (const int*)d_in[1];
  const int*   row   = ei;
  const int*   colp  = ei + NE;
  const int*   batch = (const int*)d_in[2];
  const int*   perm  = (const int*)d_in[3];
  const float* Wf    = (const float*)d_in[4];
  const float* bf    = (const float*)d_in[5];
  const float* convsW = (const float*)d_in[6];
  const float* convsB = (const float*)d_in[7];
  const float* eaW   = (const float*)d_in[8];
  const float* eaB   = (const float*)d_in[9];
  const float* naW   = (const float*)d_in[10];
  const float* naB   = (const float*)d_in[11];
  const float* ctxW  = (const float*)d_in[12];
  const float* ctxB  = (const float*)d_in[13];
  const float* objW  = (const float*)d_in[14];
  const float* objB  = (const float*)d_in[15];
  const float* fc1cW = (const float*)d_in[16];
  const float* fc1cB = (const float*)d_in[17];
  const float* fc2cW = (const float*)d_in[18];
  const float* fc2cB = (const float*)d_in[19];
  const float* fc1oW = (const float*)d_in[20];
  const float* fc1oB = (const float*)d_in[21];
  const float* fc2oW = (const float*)d_in[22];
  const float* fc2oB = (const float*)d_in[23];
  const float* fc1coW = (const float*)d_in[24];
  const float* fc1coB = (const float*)d_in[25];
  const float* fc2coW = (const float*)d_in[26];
  const float* fc2coB = (const float*)d_in[27];
  float* out = (float*)d_out;

  // ---- workspace carving (f32) ----
  float* ws = (float*)d_ws;
  size_t off = 0;
  auto alloc = [&](size_t nf) { float* p = ws + off; off += (nf + 3) & ~(size_t)3; return p; };
  float* p0 = alloc((size_t)NN * HID);
  float* p1 = alloc((size_t)NN * HID);
  float* p2 = alloc((size_t)NN * HID);
  float* p3 = alloc((size_t)NN * HID);
  float* bnx   = alloc((size_t)NN * FIN);
  float* stats = alloc(2 * HID);
  float* deg   = alloc(NN);
  float* selfn = alloc(NN);
  float* nrm   = alloc(NE);
  float* ea0   = alloc(NE);
  float* ea1   = alloc(NE);
  float* u0 = alloc(NN); float* u1 = alloc(NN);
  float* v0 = alloc(NN); float* v1 = alloc(NN);
  float* na0 = alloc(NN); float* na1 = alloc(NN);
  float* pc  = alloc((size_t)NG * HID);
  float* po  = alloc((size_t)NG * HID);
  float* pco = alloc((size_t)NG * HID);
  float* pb  = alloc((size_t)NG_PAD * HID);   // padded: WMMA A operand (M=500)
  float* pr  = alloc((size_t)NG_PAD * HID);

  auto cdiv = [](size_t a, size_t b) { return (int)((a + b - 1) / b); };

  auto bn = [&](const float* in, float* o, int n, int f) {
    hipMemsetAsync(stats, 0, 2 * (size_t)f * sizeof(float), stream);
    colsum_kernel<<<256, f, 0, stream>>>(in, n, f, stats);
    bn_apply_kernel<<<cdiv((size_t)n * f, 256), 256, 0, stream>>>(in, o, n, f, stats);
  };
  auto gemm = [&](const float* A, const float* B, const float* bias, float* C,
                  int M, int K, int fuse) {
    size_t waves = (size_t)((M + 15) / 16) * (HID / 32);
    int blocks = cdiv(waves * 32, 256);
    if (K == FIN) wmma_gemm_f32<FIN><<<blocks, 256, 0, stream>>>(A, B, bias, C, M, fuse);
    else          wmma_gemm_f32<HID><<<blocks, 256, 0, stream>>>(A, B, bias, C, M, fuse);
  };
  auto norms = [&](const float* w) {
    fill_kernel<<<cdiv(NN, 256), 256, 0, stream>>>(deg, 1.f, NN);
    deg_accum_kernel<<<cdiv(NE, 256), 256, 0, stream>>>(row, w, deg, NE);
    dis_selfn_kernel<<<cdiv(NN, 256), 256, 0, stream>>>(deg, selfn, NN);
    edge_norm_kernel<<<cdiv(NE, 256), 256, 0, stream>>>(row, colp, w, deg, nrm, NE);
  };
  auto prop = [&](const float* h, float* o) {
    prop_init_kernel<<<cdiv((size_t)NN * (HID / 4), 256), 256, 0, stream>>>(h, selfn, o);
    prop_edge_kernel<<<cdiv((size_t)NE * (HID / 4), 256), 256, 0, stream>>>(row, colp, nrm, h, o);
  };
  auto readout = [&](const float* pin, const float* W1, const float* b1,
                     const float* W2, const float* b2, float* o) {
    bn(pin, pb, NG, HID);
    gemm(pb, W1, b1, pr, NG, HID, 2);            // fc1 + bias + relu
    bn(pr, pb, NG, HID);
    fc2_logsoftmax_kernel<<<cdiv((size_t)NG * 32, 256), 256, 0, stream>>>(pb, W2, b2, o);
  };

  // ---- stage 1: bn(x); relu(x @ Wf + bf) ----
  bn(x_in, bnx, NN, FIN);
  gemm(bnx, Wf, bf, p0, NN, FIN, 2);

  // ---- shared unit-weight gcn norm ----
  norms(nullptr);

  // ---- 3 GCN layers: x = relu(prop(bn(x) @ W) + b) ----
  float* xcur = p0;
  float* alt  = p3;
  for (int i = 0; i < NLAYERS; ++i) {
    bn(xcur, p1, NN, HID);
    gemm(p1, convsW + (size_t)i * HID * HID, nullptr, p2, NN, HID, 0);
    prop(p2, alt);
    bias_relu_kernel<<<cdiv((size_t)NN * HID, 256), 256, 0, stream>>>(alt, convsB + (size_t)i * HID, NN);
    float* t = xcur; xcur = alt; alt = t;
  }

  // ---- attention (factored): u,v per node; per-edge softmax ----
  node_att_kernel<<<cdiv((size_t)NN * 32, 256), 256, 0, stream>>>(
      xcur, naW, naB, eaW, na0, na1, u0, u1, v0, v1);
  edge_att_kernel<<<cdiv(NE, 256), 256, 0, stream>>>(row, colp, u0, u1, v0, v1, eaB, ea0, ea1);
  scale_split_kernel<<<cdiv((size_t)NN * HID, 256), 256, 0, stream>>>(xcur, na0, na1, p1, p2);

  // ---- ctx branch: xc(p1) ----
  bn(p1, alt, NN, HID);
  gemm(alt, ctxW, nullptr, p1, NN, HID, 0);
  norms(ea0);
  prop(p1, alt);
  bias_relu_kernel<<<cdiv((size_t)NN * HID, 256), 256, 0, stream>>>(alt, ctxB, NN);
  hipMemsetAsync(pc, 0, (size_t)NG * HID * sizeof(float), stream);
  pool_kernel<<<cdiv((size_t)NN * (HID / 4), 256), 256, 0, stream>>>(alt, batch, pc);

  // ---- obj branch: xo(p2) ----
  bn(p2, p1, NN, HID);
  gemm(p1, objW, nullptr, p2, NN, HID, 0);
  norms(ea1);
  prop(p2, p1);
  bias_relu_kernel<<<cdiv((size_t)NN * HID, 256), 256, 0, stream>>>(p1, objB, NN);
  hipMemsetAsync(po, 0, (size_t)NG * HID * sizeof(float), stream);
  pool_kernel<<<cdiv((size_t)NN * (HID / 4), 256), 256, 0, stream>>>(p1, batch, po);

  // ---- readouts ----
  readout(pc, fc1cW, fc1cB, fc2cW, fc2cB, out);
  readout(po, fc1oW, fc1oB, fc2oW, fc2oB, out + NG * NCLS);
  perm_add_kernel<<<cdiv((size_t)NG * HID, 256), 256, 0, stream>>>(pc, po, perm, pco);
  readout(pco, fc1coW, fc1coB, fc2coW, fc2coB, out + 2 * NG * NCLS);
}